// TopKPooling_20796231647786
// MI455X (gfx1250) — compile-verified
//
#include <hip/hip_runtime.h>
#include <hip/hip_bf16.h>
#include <stdint.h>

// TopK (k=64, descending, values only) over rows of 16384 floats.
// B=32, C=256, H*W=16384 -> 8192 rows. Memory-bound: 512MB read once
// (~22us floor at 23.3 TB/s). Strategy: async-stage each 64KB row into LDS
// (CDNA5 GLOBAL_LOAD_ASYNC_TO_LDS_B128), 4-pass 8-bit radix select in LDS,
// collect 64 winners, rank-sort, store.

#define ROWLEN  16384
#define K       64
#define THREADS 256   // 8 wave32 per workgroup

#if defined(__HIP_DEVICE_COMPILE__)
#  if __has_builtin(__builtin_amdgcn_global_load_async_to_lds_b128) && \
      __has_builtin(__builtin_amdgcn_s_wait_asynccnt)
#    define USE_ASYNC_LDS 1
#  endif
#endif

#ifdef USE_ASYNC_LDS
// Match the builtin's expected parameter types exactly:
//   param0: int __attribute__((vector_size(16))) * in AS1 (global/device)
//   param1: same vector type in AS3 (LDS/shared)
typedef int vint4_t __attribute__((vector_size(16)));
typedef __attribute__((address_space(1))) vint4_t* gas_vint4_p;
typedef __attribute__((address_space(3))) vint4_t* las_vint4_p;
#endif

__global__ __launch_bounds__(THREADS)
void topk64_kernel(const float* __restrict__ x, float* __restrict__ out, int rows) {
    extern __shared__ unsigned sdata[];        // ROWLEN words = 64 KB dynamic LDS
    __shared__ unsigned hist[256];
    __shared__ unsigned cand[K];
    __shared__ unsigned s_prefix, s_rem, s_nG, s_nE;

    const int row = blockIdx.x;                // uniform per block
    if (row >= rows) return;
    const float* src = x + (size_t)row * ROWLEN;
    const int tid = threadIdx.x;

    // ---- Stage row: global -> LDS (read HBM exactly once) ----
#ifdef USE_ASYNC_LDS
    {
        const char* gbase = (const char*)src;
        char* lbase = (char*)sdata;
        // 16384 floats = 4096 x 16B chunks; 256 threads x 16 iterations
        for (int i = 0; i < (ROWLEN / 4) / THREADS; ++i) {
            const size_t chunk = (size_t)(i * THREADS + tid) * 16;
            __builtin_amdgcn_global_load_async_to_lds_b128(
                (gas_vint4_p)(gbase + chunk),
                (las_vint4_p)(lbase + chunk),
                0, 0);
        }
        __builtin_amdgcn_s_wait_asynccnt(0);   // this wave's async loads done
    }
#else
    {
        const uint4* g = (const uint4*)src;
        uint4* l = (uint4*)sdata;
        for (int i = tid; i < ROWLEN / 4; i += THREADS) l[i] = g[i];
    }
#endif
    if (tid == 0) { s_prefix = 0u; s_rem = (unsigned)K; }
    hist[tid] = 0u;
    __syncthreads();                            // all waves' loads visible

    // ---- Pass 0: map float bits -> order-preserving u32, histogram top byte
    for (int i = tid; i < ROWLEN; i += THREADS) {
        unsigned u = sdata[i];
        u = (u & 0x80000000u) ? ~u : (u | 0x80000000u);
        sdata[i] = u;
        atomicAdd(&hist[u >> 24], 1u);
    }
    __syncthreads();
    if (tid == 0) {
        unsigned rem = s_rem; int bin = 255;
        for (; bin > 0; --bin) { unsigned c = hist[bin]; if (c >= rem) break; rem -= c; }
        s_prefix = (s_prefix << 8) | (unsigned)bin;
        s_rem = rem;
    }
    __syncthreads();

    // ---- Passes 1..3: refine next 8 bits among prefix-matching elements
    for (int pass = 1; pass < 4; ++pass) {
        hist[tid] = 0u;
        __syncthreads();
        const unsigned prefix = s_prefix;
        const int shiftHi = 32 - 8 * pass;
        const int shift   = 24 - 8 * pass;
        for (int i = tid; i < ROWLEN; i += THREADS) {
            const unsigned u = sdata[i];
            if ((u >> shiftHi) == prefix)
                atomicAdd(&hist[(u >> shift) & 255u], 1u);
        }
        __syncthreads();
        if (tid == 0) {
            unsigned rem = s_rem; int bin = 255;
            for (; bin > 0; --bin) { unsigned c = hist[bin]; if (c >= rem) break; rem -= c; }
            s_prefix = (s_prefix << 8) | (unsigned)bin;
            s_rem = rem;
        }
        __syncthreads();
    }

    const unsigned T      = s_prefix;  // exact mapped value of the k-th element
    const unsigned needEq = s_rem;     // how many ties of T belong in top-k
    if (tid == 0) { s_nG = 0u; s_nE = 0u; }
    __syncthreads();

    // ---- Collect the 64 winners
    for (int i = tid; i < ROWLEN; i += THREADS) {
        const unsigned u = sdata[i];
        if (u > T) {
            const unsigned p = atomicAdd(&s_nG, 1u);   // p < K - needEq by construction
            if (p < (unsigned)K) cand[p] = u;
        } else if (u == T) {
            const unsigned e = atomicAdd(&s_nE, 1u);
            if (e < needEq) cand[(K - needEq) + e] = u;
        }
    }
    __syncthreads();

    // ---- Rank-sort 64 values descending, unmap, store
    if (tid < K) {
        const unsigned v = cand[tid];
        int rank = 0;
        for (int j = 0; j < K; ++j) {
            const unsigned w = cand[j];
            rank += (w > v || (w == v && j < tid)) ? 1 : 0;
        }
        const unsigned b = (v & 0x80000000u) ? (v ^ 0x80000000u) : ~v;
        out[(size_t)row * K + rank] = __uint_as_float(b);
    }
}

extern "C" void kernel_launch(void* const* d_in, const int* in_sizes, int n_in,
                              void* d_out, int out_size, void* d_ws, size_t ws_size,
                              hipStream_t stream) {
    const float* x = (const float*)d_in[0];   // [B, C, H, W] float32
    // d_in[1] is k == 64 (compile-time constant K here)
    float* out = (float*)d_out;               // [B, C, 64] float32
    const int rows = in_sizes[0] / ROWLEN;    // 8192
    const size_t shmem = (size_t)ROWLEN * sizeof(unsigned);  // 64 KB dynamic LDS
    hipLaunchKernelGGL(topk64_kernel, dim3(rows), dim3(THREADS), shmem, stream,
                       x, out, rows);
}